// DeepSetsAggregator_3393024164426
// MI455X (gfx1250) — compile-verified
//
#include <hip/hip_runtime.h>
#include <stdint.h>

// ---------------------------------------------------------------------------
// DeepSets segment aggregator (sum | mean | max) for sorted segment ids.
// Memory-bound: 512MB stream of x -> ~22us floor at 23.3 TB/s HBM.
// Uses CDNA5 async global->LDS copies (ASYNCcnt) with a 4-deep ring buffer
// per wave to keep the HBM pipe full.
// ---------------------------------------------------------------------------

#define SEG_H   128          // feature dim
#define WAVES   8            // waves per block (wave32)
#define BLOCK_T (WAVES * 32) // 256 threads
#define DEPTH   4            // ring-buffer depth (power of two)

// One wave copies one full 512B row: 32 lanes x b128 (16B) each.
__device__ __forceinline__ void async_row_b128(const float* g, uint32_t lds_addr) {
  // VDST = LDS byte address VGPR, VADDR = 64-bit global address VGPR pair.
  asm volatile("global_load_async_to_lds_b128 %0, %1, off"
               :: "v"(lds_addr), "v"(g)
               : "memory");
}

#define WAIT_ASYNC(n) asm volatile("s_wait_asynccnt " #n ::: "memory")

// ---------------------------------------------------------------------------
// Kernel 1: segment boundaries via binary search over the sorted batch array.
// starts[b] = lower_bound(batch, b); starts[nseg] = n (falls out naturally).
// ---------------------------------------------------------------------------
__global__ void seg_bounds_kernel(const int* __restrict__ batch, int n, int nseg,
                                  int* __restrict__ starts) {
  int b = blockIdx.x * blockDim.x + threadIdx.x;
  if (b > nseg) return;
  int lo = 0, hi = n;
  while (lo < hi) {
    int mid = (lo + hi) >> 1;
    if (batch[mid] < b) lo = mid + 1; else hi = mid;
  }
  starts[b] = lo;
}

// ---------------------------------------------------------------------------
// Kernel 2: one block per segment. Wave w handles rows start+w, start+w+8, ...
// streamed through LDS via async copies; cross-wave reduce in LDS at the end.
// ---------------------------------------------------------------------------
__global__ __launch_bounds__(BLOCK_T)
void seg_agg_kernel(const float* __restrict__ x, const int* __restrict__ starts,
                    float* __restrict__ out) {
  __shared__ __align__(16) float stage[WAVES][DEPTH][SEG_H]; // 16 KB
  __shared__ float psum[WAVES * SEG_H];                      // 4 KB
  __shared__ float pmax[WAVES * SEG_H];                      // 4 KB

  const int b    = blockIdx.x;
  const int s    = starts[b];
  const int e    = starts[b + 1];
  const int tid  = threadIdx.x;
  const int wave = tid >> 5;
  const int lane = tid & 31;

  const int r0    = s + wave;                                    // wave-uniform
  const int total = (e > r0) ? ((e - r0 + WAVES - 1) / WAVES) : 0;

  const float*   g0   = x + (size_t)r0 * SEG_H + (lane << 2);
  const uint32_t lds0 = (uint32_t)(uintptr_t)&stage[wave][0][lane << 2];

  float sx = 0.f, sy = 0.f, sz = 0.f, sw = 0.f;
  const float NEG = -__builtin_inff();
  float mx = NEG, my = NEG, mz = NEG, mw = NEG;

  // Prologue: keep up to DEPTH-1 = 3 rows in flight per wave.
  if (0 < total) async_row_b128(g0,                              lds0);
  if (1 < total) async_row_b128(g0 + (size_t)1 * WAVES * SEG_H,  lds0 + 1u * SEG_H * 4);
  if (2 < total) async_row_b128(g0 + (size_t)2 * WAVES * SEG_H,  lds0 + 2u * SEG_H * 4);

  for (int k = 0; k < total; ++k) {
    if (k + 3 < total) {
      // Slot (k+3)&3 was consumed at iteration k-1; safe to overwrite.
      async_row_b128(g0 + (size_t)(k + 3) * WAVES * SEG_H,
                     lds0 + (uint32_t)(((k + 3) & (DEPTH - 1)) * SEG_H * 4));
      WAIT_ASYNC(3);              // async loads complete in order -> row k resident
    } else {
      const int rem = total - k - 1;  // rows still in flight after this one
      if      (rem == 2) WAIT_ASYNC(2);
      else if (rem == 1) WAIT_ASYNC(1);
      else               WAIT_ASYNC(0);
    }
    const float4 v = *(const float4*)&stage[wave][k & (DEPTH - 1)][lane << 2];
    sx += v.x; sy += v.y; sz += v.z; sw += v.w;
    mx = fmaxf(mx, v.x); my = fmaxf(my, v.y);
    mz = fmaxf(mz, v.z); mw = fmaxf(mw, v.w);
  }

  // Per-wave partials -> LDS (each (wave,lane) writes a unique float4).
  const int pb = wave * SEG_H + (lane << 2);
  psum[pb + 0] = sx; psum[pb + 1] = sy; psum[pb + 2] = sz; psum[pb + 3] = sw;
  pmax[pb + 0] = mx; pmax[pb + 1] = my; pmax[pb + 2] = mz; pmax[pb + 3] = mw;
  __syncthreads();

  // Cross-wave reduce: thread f (< 128) owns feature f. Coalesced stores.
  if (tid < SEG_H) {
    float ssum = 0.f, smax = NEG;
#pragma unroll
    for (int w = 0; w < WAVES; ++w) {
      ssum += psum[w * SEG_H + tid];
      smax  = fmaxf(smax, pmax[w * SEG_H + tid]);
    }
    const int   cnt = e - s;
    const size_t ob = (size_t)b * (3 * SEG_H);
    out[ob + tid]             = ssum;
    out[ob + SEG_H + tid]     = ssum / fmaxf((float)cnt, 1.0f);
    out[ob + 2 * SEG_H + tid] = (cnt > 0) ? smax : 0.0f;   // PyG scatter-max fill
  }
}

// ---------------------------------------------------------------------------
// Launch: inputs are (x[N,H] f32, batch[N] i32, batch_size scalar on device).
// N from in_sizes[1]; B from out_size / (3*H). Workspace holds starts[B+1].
// ---------------------------------------------------------------------------
extern "C" void kernel_launch(void* const* d_in, const int* in_sizes, int n_in,
                              void* d_out, int out_size, void* d_ws, size_t ws_size,
                              hipStream_t stream) {
  const float* x     = (const float*)d_in[0];
  const int*   batch = (const int*)d_in[1];
  const int    n     = in_sizes[1];
  const int    nseg  = out_size / (3 * SEG_H);

  int* starts = (int*)d_ws;  // (nseg + 1) ints

  const int bt = 256;
  const int bb = (nseg + 1 + bt - 1) / bt;
  seg_bounds_kernel<<<bb, bt, 0, stream>>>(batch, n, nseg, starts);
  seg_agg_kernel<<<nseg, BLOCK_T, 0, stream>>>(x, starts, (float*)d_out);
}